// OKRRouter_11536282157085
// MI455X (gfx1250) — compile-verified
//
#include <hip/hip_runtime.h>

typedef __attribute__((ext_vector_type(2))) float v2f;
typedef __attribute__((ext_vector_type(4))) float v4f;
typedef __attribute__((ext_vector_type(8))) float v8f;

#define DDIM 2048
#define EDIM 64
#define NTOK 16384
#define EPS  1.5f
#define NEG_INF -1000000000.0f
#define LDS_STRIDE 132   // 128 cols + 4 pad -> bank stride 4 for column reads

// ---------------------------------------------------------------------------
// Kernel 1: transpose secret [DDIM, EDIM] -> secretT [EDIM, DDIM] via LDS tile
// so the main kernel's B loads are contiguous b128 for BOTH projections.
// ---------------------------------------------------------------------------
__global__ __launch_bounds__(256) void transpose_secret_kernel(
    const float* __restrict__ in,    // [DDIM, EDIM]
    float* __restrict__ outT)        // [EDIM, DDIM]
{
    __shared__ float t[64][65];      // 65 stride -> conflict-free transpose
    const int kb = blockIdx.x * 64;  // 32 blocks cover DDIM
    const int c  = threadIdx.x & 63;
    const int r0 = threadIdx.x >> 6; // 0..3

    #pragma unroll
    for (int r = r0; r < 64; r += 4)
        t[r][c] = in[(size_t)(kb + r) * EDIM + c];   // coalesced read

    __syncthreads();

    #pragma unroll
    for (int r = r0; r < 64; r += 4)                 // coalesced write
        outT[(size_t)r * DDIM + kb + c] = t[c][r];   // bank-conflict-free
}

// ---------------------------------------------------------------------------
// Kernel 2: fused dual-GEMM router.
// One block = 16 tokens, 8 waves. Waves 0..3: gate logits (cols 0..63),
// waves 4..7: watermark bias (cols 64..127). Each wave accumulates a 16x16
// fp32 tile with V_WMMA_F32_16X16X4_F32 (exact fp32 - the eps-mask / top-2
// decisions are precision-sensitive).
// ---------------------------------------------------------------------------
__global__ __launch_bounds__(256) void okr_router_kernel(
    const float* __restrict__ hidden,   // [NTOK, DDIM]
    const float* __restrict__ gateW,    // [EDIM, DDIM]
    const float* __restrict__ secret,   // [DDIM, EDIM] (fallback path)
    const float* __restrict__ secretT,  // [EDIM, DDIM] or nullptr
    float* __restrict__ out)            // [2*NTOK weights][2*NTOK expert ids]
{
    __shared__ float lds[16 * LDS_STRIDE];

    const int tid  = threadIdx.x;
    const int wave = tid >> 5;
    const int lane = tid & 31;
    const int l15  = lane & 15;
    const int half = lane >> 4;   // lanes 0-15: K 0..3 of each 8-chunk, 16-31: K 4..7
    const int tile = blockIdx.x;

    v8f acc = {};
    const float* pA = hidden + (size_t)(tile * 16 + l15) * DDIM + 4 * half;

    if (secretT != nullptr || wave < 4) {
        // Uniform path: B rows are contiguous over K for gate and secretT.
        const float* base = (wave < 4) ? gateW : secretT;
        const float* pB = base + (size_t)((wave & 3) * 16 + l15) * DDIM + 4 * half;
        #pragma unroll 4
        for (int c = 0; c < DDIM; c += 8) {
            v4f a = *(const v4f*)(pA + c);
            v4f b = *(const v4f*)(pB + c);
            v2f a0 = {a.x, a.y}, a1 = {a.z, a.w};
            v2f b0 = {b.x, b.y}, b1 = {b.z, b.w};
            acc = __builtin_amdgcn_wmma_f32_16x16x4_f32(
                false, a0, false, b0, (short)0, acc, false, false);
            acc = __builtin_amdgcn_wmma_f32_16x16x4_f32(
                false, a1, false, b1, (short)0, acc, false, false);
        }
    } else {
        // Fallback (no workspace): strided b32 loads from secret [DDIM, EDIM].
        const int se = (wave - 4) * 16 + l15;
        const float* pB = secret + se + (size_t)(4 * half) * EDIM;
        #pragma unroll 4
        for (int c = 0; c < DDIM; c += 8) {
            v4f a = *(const v4f*)(pA + c);
            const float* pk = pB + (size_t)c * EDIM;
            v2f b0 = {pk[0],        pk[EDIM]};
            v2f b1 = {pk[2 * EDIM], pk[3 * EDIM]};
            v2f a0 = {a.x, a.y}, a1 = {a.z, a.w};
            acc = __builtin_amdgcn_wmma_f32_16x16x4_f32(
                false, a0, false, b0, (short)0, acc, false, false);
            acc = __builtin_amdgcn_wmma_f32_16x16x4_f32(
                false, a1, false, b1, (short)0, acc, false, false);
        }
    }

    // C/D layout: VGPR r = row r (lanes 0-15) / row 8+r (lanes 16-31), col = l15.
    const int cb   = wave * 16;     // gate waves -> cols 0..63, secret -> 64..127
    const int row0 = half * 8;
    #pragma unroll
    for (int r = 0; r < 8; ++r)
        lds[(row0 + r) * LDS_STRIDE + cb + l15] = acc[r];

    __syncthreads();

    // Epilogue: one thread per token.
    if (tid < 16) {
        const float* Lg = &lds[tid * LDS_STRIDE];      // raw gate logits [64]
        const float* Wb = Lg + 64;                     // watermark bias  [64]

        float m = Lg[0];
        #pragma unroll 8
        for (int e = 1; e < EDIM; ++e) m = fmaxf(m, Lg[e]);
        const float thr = m - EPS;

        // top-2 of masked bias; strict > keeps lowest index on ties (lax.top_k)
        float s1 = -3.4e38f, s2 = -3.4e38f;
        int   i1 = 0,        i2 = 0;
        for (int e = 0; e < EDIM; ++e) {
            float s = (Lg[e] >= thr) ? Wb[e] : NEG_INF;
            if (s > s1)      { s2 = s1; i2 = i1; s1 = s; i1 = e; }
            else if (s > s2) { s2 = s;  i2 = e; }
        }

        const float l1 = Lg[i1], l2 = Lg[i2];
        const float mx = fmaxf(l1, l2);
        float w1 = __expf(l1 - mx);
        float w2 = __expf(l2 - mx);
        const float inv = 1.0f / (w1 + w2);
        w1 *= inv; w2 *= inv;

        const int token = tile * 16 + tid;
        out[token * 2 + 0] = w1;
        out[token * 2 + 1] = w2;
        out[2 * NTOK + token * 2 + 0] = (float)i1;
        out[2 * NTOK + token * 2 + 1] = (float)i2;
    }
}

extern "C" void kernel_launch(void* const* d_in, const int* in_sizes, int n_in,
                              void* d_out, int out_size, void* d_ws, size_t ws_size,
                              hipStream_t stream) {
    const float* hidden = (const float*)d_in[0];   // [4,4096,2048]
    const float* gateW  = (const float*)d_in[1];   // [64,2048]
    const float* secret = (const float*)d_in[2];   // [2048,64]
    // d_in[3] = top_k (== 2), baked into the kernel
    (void)in_sizes; (void)n_in; (void)out_size;

    const size_t needT = (size_t)EDIM * DDIM * sizeof(float);   // 512 KB
    float* secretT = nullptr;
    if (ws_size >= needT) {
        secretT = (float*)d_ws;
        transpose_secret_kernel<<<DDIM / 64, 256, 0, stream>>>(secret, secretT);
    }

    okr_router_kernel<<<NTOK / 16, 256, 0, stream>>>(hidden, gateW, secret,
                                                     secretT, (float*)d_out);
}